// NN_Custom_torch_35768487641278
// MI455X (gfx1250) — compile-verified
//
#include <hip/hip_runtime.h>
#include <math.h>

#define NDIM 4096
#define KSPLIT 8
#define KCH (NDIM / KSPLIT)      // 512 columns per K-split
#define KSTAGE 64                // columns per pipeline stage
#define NSTAGE (KCH / KSTAGE)    // 8 stages
#define LROW (KSTAGE + 4)        // 68 floats: 16B-aligned stride, conflict-free reads
#define WAVES 8
#define ROWS_PER_WAVE 16
#define ROWS_PER_BLOCK (WAVES * ROWS_PER_WAVE)  // 128
#define LN_EPS 1e-5f

typedef __attribute__((ext_vector_type(2))) float v2f;
typedef __attribute__((ext_vector_type(8))) float v8f;

// ---------------------------------------------------------------------------
// i_out = input_raw + b_i   (reference: LN(zeros)=beta, then += input)
// ---------------------------------------------------------------------------
__global__ __launch_bounds__(256) void prep_kernel(const float* __restrict__ input_raw,
                                                   const float* __restrict__ b_i,
                                                   float* __restrict__ i_out) {
  const int t = threadIdx.x;
#pragma unroll
  for (int j = 0; j < NDIM / 256; ++j) {
    const int idx = t + j * 256;
    i_out[idx] = input_raw[idx] + b_i[idx];
  }
}

// ---------------------------------------------------------------------------
// GEMV via V_WMMA_F32_16X16X4_F32 with async global->LDS double buffering.
//   grid = (NDIM/128, KSPLIT), block = 256 (8 waves).
//   Wave w handles rows r0=(bx*8+w)*16 over K range [ky*512, +512).
//   Staging: per stage, each wave pulls its 16 rows x 64 cols tile into its
//   private LDS buffer with 8x GLOBAL_LOAD_ASYNC_TO_LDS_B128 (each half-wave
//   reads a contiguous 256B row segment -> fully coalesced), double-buffered
//   against the WMMA compute; sync via s_wait_asynccnt only (no barriers).
//   A fragment (16x4 f32, documented layout) read from LDS:
//     lanes 0-15 (row m):  W[r][k], W[r][k+1]; lanes 16-31: W[r][k+2], W[r][k+3]
//   B fragment: x[k..k+3] broadcast to all 16 columns (mirrors A's K striping).
//   D: all columns equal -> lanes 0/16 hold y-partials for rows r0+v / r0+8+v.
// ---------------------------------------------------------------------------
__global__ __launch_bounds__(256) void gemv_wmma_kernel(const float* __restrict__ W,
                                                        const float* __restrict__ x,
                                                        float* __restrict__ Upart) {
  __shared__ float xs[KCH];
  __shared__ float wbuf[WAVES][2][ROWS_PER_WAVE * LROW];

  const int ky = blockIdx.y;
  const int k0 = ky * KCH;
  const int t = threadIdx.x;

  // stage this block's x chunk into LDS (coalesced, 2 elems/thread)
  xs[t] = x[k0 + t];
  xs[t + 256] = x[k0 + t + 256];
  __syncthreads();

  const int wave = t >> 5;
  const int lane = t & 31;
  const int half = lane >> 4;          // compute: 0 -> K+{0,1}, 1 -> K+{2,3}
  const int m    = lane & 15;          // compute: row within 16-row tile
  const int row0 = blockIdx.x * ROWS_PER_BLOCK + wave * ROWS_PER_WAVE;

  // load mapping: instruction i covers rows {2i, 2i+1}; lane -> (row parity, 16B seg)
  const int lrow = lane >> 4;          // row parity within the pair
  const int lseg = lane & 15;          // which 16B segment of the 256B row chunk
  const uint64_t gbase = (uint64_t)(uintptr_t)W +
      ((uint64_t)(row0 + lrow) * NDIM + (uint64_t)k0 + (uint64_t)(lseg * 4)) * 4ull;
  const uint32_t lb0 =
      (uint32_t)(uintptr_t)&wbuf[wave][0][lrow * LROW + lseg * 4];
  const uint32_t lb1 =
      (uint32_t)(uintptr_t)&wbuf[wave][1][lrow * LROW + lseg * 4];

  auto issue_stage = [&](int s) {
    const uint64_t gs = gbase + (uint64_t)s * (KSTAGE * 4ull);
    const uint32_t lb = (s & 1) ? lb1 : lb0;
#pragma unroll
    for (int i = 0; i < 8; ++i) {
      const uint64_t ga = gs + (uint64_t)(2 * i) * (uint64_t)(NDIM * 4);
      const uint32_t la = lb + (uint32_t)(2 * i) * (uint32_t)(LROW * 4);
      asm volatile("global_load_async_to_lds_b128 %0, %1, off"
                   :: "v"(la), "v"(ga)
                   : "memory");
    }
  };

  issue_stage(0);

  v8f c = {};
#pragma unroll
  for (int s = 0; s < NSTAGE; ++s) {
    if (s + 1 < NSTAGE) {
      issue_stage(s + 1);
      // stage s's 8 loads are the oldest; <=8 outstanding means they're done
      asm volatile("s_wait_asynccnt 0x8" ::: "memory");
    } else {
      asm volatile("s_wait_asynccnt 0x0" ::: "memory");
    }
    const float* wb = &wbuf[wave][s & 1][0];
#pragma unroll
    for (int kk = 0; kk < KSTAGE; kk += 4) {
      v2f a = *(const v2f*)(wb + m * LROW + kk + 2 * half);        // ds_load_b64
      v2f b = *(const v2f*)(xs + s * KSTAGE + kk + 2 * half);      // broadcast
      c = __builtin_amdgcn_wmma_f32_16x16x4_f32(false, a, false, b, (short)0, c,
                                                false, false);
    }
  }

  // every column of D is identical; lane 0 carries rows r0..r0+7 (VGPR v -> row
  // r0+v), lane 16 carries rows r0+8..r0+15.
  if (lane == 0 || lane == 16) {
    float* yp = Upart + (size_t)ky * NDIM + row0 + half * 8;
#pragma unroll
    for (int v = 0; v < 8; ++v) yp[v] = c[v];
  }
}

// ---------------------------------------------------------------------------
// Single-block (256-thread) LayerNorm statistics over 4096 elements.
// ---------------------------------------------------------------------------
__device__ __forceinline__ void block_ln_stats(float s, float q, float* red,
                                               float& mu, float& rstd) {
  const int t = threadIdx.x;
  red[t] = s;
  red[256 + t] = q;
  __syncthreads();
#pragma unroll
  for (int off = 128; off > 0; off >>= 1) {
    if (t < off) {
      red[t] += red[t + off];
      red[256 + t] += red[256 + t + off];
    }
    __syncthreads();
  }
  const float m = red[0] * (1.0f / NDIM);
  const float v = red[256] * (1.0f / NDIM) - m * m;
  mu = m;
  rstd = rsqrtf(v + LN_EPS);
  __syncthreads();  // allow lds reuse by the next reduction
}

// ---------------------------------------------------------------------------
// Phase 2 fused kernel (single block):
//   h1_fb = LN(sum U1; g_h1, bln_h1)
//   h2_fb = LN(sum U2; g_h2, bln_h2)
//   h1_p2 = relu(LN(sum U3 + h1_fb + bias_h1; g_h1, bln_h1))
//   h2_p2 = relu(LN(h2_fb + bias_h2; g_h2, bln_h2))
// ---------------------------------------------------------------------------
__global__ __launch_bounds__(256) void phase2_kernel(
    const float* __restrict__ U1, const float* __restrict__ U2,
    const float* __restrict__ U3,
    const float* __restrict__ g_h1, const float* __restrict__ bln_h1,
    const float* __restrict__ g_h2, const float* __restrict__ bln_h2,
    const float* __restrict__ bias_h1, const float* __restrict__ bias_h2,
    float* __restrict__ h1_p2, float* __restrict__ h2_p2) {
  __shared__ float red[512];
  const int t = threadIdx.x;
  float a1[16], a2[16], a3[16], f1[16], f2[16];
  float s1 = 0.f, q1 = 0.f, s2 = 0.f, q2 = 0.f;

#pragma unroll
  for (int j = 0; j < 16; ++j) {
    const int idx = t + j * 256;
    float u1 = 0.f, u2 = 0.f, u3 = 0.f;
#pragma unroll
    for (int p = 0; p < KSPLIT; ++p) {
      u1 += U1[p * NDIM + idx];
      u2 += U2[p * NDIM + idx];
      u3 += U3[p * NDIM + idx];
    }
    a1[j] = u1; a2[j] = u2; a3[j] = u3;
    s1 += u1; q1 += u1 * u1;
    s2 += u2; q2 += u2 * u2;
  }

  float mu, rstd;
  block_ln_stats(s1, q1, red, mu, rstd);
#pragma unroll
  for (int j = 0; j < 16; ++j) {
    const int idx = t + j * 256;
    f1[j] = (a1[j] - mu) * rstd * g_h1[idx] + bln_h1[idx];  // h1_fb
  }

  block_ln_stats(s2, q2, red, mu, rstd);
#pragma unroll
  for (int j = 0; j < 16; ++j) {
    const int idx = t + j * 256;
    f2[j] = (a2[j] - mu) * rstd * g_h2[idx] + bln_h2[idx];  // h2_fb
  }

  float s3 = 0.f, q3 = 0.f;
#pragma unroll
  for (int j = 0; j < 16; ++j) {
    const int idx = t + j * 256;
    const float v = a3[j] + f1[j] + bias_h1[idx];
    a3[j] = v; s3 += v; q3 += v * v;
  }
  block_ln_stats(s3, q3, red, mu, rstd);
#pragma unroll
  for (int j = 0; j < 16; ++j) {
    const int idx = t + j * 256;
    const float v = (a3[j] - mu) * rstd * g_h1[idx] + bln_h1[idx];
    h1_p2[idx] = fmaxf(v, 0.f);
  }

  float s4 = 0.f, q4 = 0.f;
#pragma unroll
  for (int j = 0; j < 16; ++j) {
    const int idx = t + j * 256;
    const float v = f2[j] + bias_h2[idx];
    f2[j] = v; s4 += v; q4 += v * v;
  }
  block_ln_stats(s4, q4, red, mu, rstd);
#pragma unroll
  for (int j = 0; j < 16; ++j) {
    const int idx = t + j * 256;
    const float v = (f2[j] - mu) * rstd * g_h2[idx] + bln_h2[idx];
    h2_p2[idx] = fmaxf(v, 0.f);
  }
}

// ---------------------------------------------------------------------------
// out = relu(LN(sum_p Upart[p] + add; g, b))   (single block)
// ---------------------------------------------------------------------------
__global__ __launch_bounds__(256) void ln_relu_kernel(
    const float* __restrict__ Upart, const float* __restrict__ add,
    const float* __restrict__ g, const float* __restrict__ b,
    float* __restrict__ out) {
  __shared__ float red[512];
  const int t = threadIdx.x;
  float a[16];
  float s = 0.f, q = 0.f;
#pragma unroll
  for (int j = 0; j < 16; ++j) {
    const int idx = t + j * 256;
    float u = add[idx];
#pragma unroll
    for (int p = 0; p < KSPLIT; ++p) u += Upart[p * NDIM + idx];
    a[j] = u; s += u; q += u * u;
  }
  float mu, rstd;
  block_ln_stats(s, q, red, mu, rstd);
#pragma unroll
  for (int j = 0; j < 16; ++j) {
    const int idx = t + j * 256;
    const float v = (a[j] - mu) * rstd * g[idx] + b[idx];
    out[idx] = fmaxf(v, 0.f);
  }
}

// ---------------------------------------------------------------------------
// o = tanh(LN(bln_o + sum U6 + sum U7 + bias_o; g_o, bln_o))   (single block)
// ---------------------------------------------------------------------------
__global__ __launch_bounds__(256) void final_kernel(
    const float* __restrict__ U6, const float* __restrict__ U7,
    const float* __restrict__ bias_o, const float* __restrict__ g_o,
    const float* __restrict__ bln_o, float* __restrict__ out) {
  __shared__ float red[512];
  const int t = threadIdx.x;
  float a[16];
  float s = 0.f, q = 0.f;
#pragma unroll
  for (int j = 0; j < 16; ++j) {
    const int idx = t + j * 256;
    float u = bln_o[idx] + bias_o[idx];
#pragma unroll
    for (int p = 0; p < KSPLIT; ++p) u += U6[p * NDIM + idx] + U7[p * NDIM + idx];
    a[j] = u; s += u; q += u * u;
  }
  float mu, rstd;
  block_ln_stats(s, q, red, mu, rstd);
#pragma unroll
  for (int j = 0; j < 16; ++j) {
    const int idx = t + j * 256;
    const float v = (a[j] - mu) * rstd * g_o[idx] + bln_o[idx];
    out[idx] = tanhf(v);
  }
}

// ---------------------------------------------------------------------------
extern "C" void kernel_launch(void* const* d_in, const int* in_sizes, int n_in,
                              void* d_out, int out_size, void* d_ws, size_t ws_size,
                              hipStream_t stream) {
  const float* input_raw = (const float*)d_in[0];
  const float* h1_prev   = (const float*)d_in[1];
  const float* h2_prev   = (const float*)d_in[2];
  const float* W_I_H1    = (const float*)d_in[3];
  const float* W_H1_H1   = (const float*)d_in[4];
  const float* W_H1_H2   = (const float*)d_in[5];
  const float* W_H2_H1   = (const float*)d_in[6];
  const float* W_H2_H2   = (const float*)d_in[7];
  const float* W_H1_O    = (const float*)d_in[8];
  const float* W_H2_O    = (const float*)d_in[9];
  const float* bias_h1   = (const float*)d_in[10];
  const float* bias_h2   = (const float*)d_in[11];
  const float* bias_o    = (const float*)d_in[12];
  /* d_in[13] = g_i : unused (LN of zeros reduces to beta) */
  const float* b_i       = (const float*)d_in[14];
  const float* g_h1      = (const float*)d_in[15];
  const float* bln_h1    = (const float*)d_in[16];
  const float* g_h2      = (const float*)d_in[17];
  const float* bln_h2    = (const float*)d_in[18];
  const float* g_o       = (const float*)d_in[19];
  const float* bln_o     = (const float*)d_in[20];

  float* ws = (float*)d_ws;
  float* i_out = ws;                       // 4096
  float* U1 = ws + 4096;                   // KSPLIT*4096 each
  float* U2 = U1 + KSPLIT * NDIM;
  float* U3 = U2 + KSPLIT * NDIM;
  float* U4 = U3 + KSPLIT * NDIM;
  float* U5 = U4 + KSPLIT * NDIM;
  float* U6 = U5 + KSPLIT * NDIM;
  float* U7 = U6 + KSPLIT * NDIM;
  float* h1_p2 = U7 + KSPLIT * NDIM;       // 4096 each
  float* h2_p2 = h1_p2 + NDIM;
  float* h1_f  = h2_p2 + NDIM;
  float* h2_f  = h1_f + NDIM;

  const dim3 ggrid(NDIM / ROWS_PER_BLOCK, KSPLIT);
  const dim3 gblk(256);

  // Phase 1 GEMVs (each HBM-bound: streams a 64 MiB weight matrix once)
  prep_kernel<<<1, 256, 0, stream>>>(input_raw, b_i, i_out);
  gemv_wmma_kernel<<<ggrid, gblk, 0, stream>>>(W_H1_H1, h1_prev, U1);
  gemv_wmma_kernel<<<ggrid, gblk, 0, stream>>>(W_H2_H2, h2_prev, U2);
  gemv_wmma_kernel<<<ggrid, gblk, 0, stream>>>(W_I_H1, i_out, U3);

  // Phase 2: fused feedback LNs + activations
  phase2_kernel<<<1, 256, 0, stream>>>(U1, U2, U3, g_h1, bln_h1, g_h2, bln_h2,
                                       bias_h1, bias_h2, h1_p2, h2_p2);

  // Phase 3 GEMVs
  gemv_wmma_kernel<<<ggrid, gblk, 0, stream>>>(W_H2_H1, h2_p2, U4);
  gemv_wmma_kernel<<<ggrid, gblk, 0, stream>>>(W_H1_H2, h1_p2, U5);
  ln_relu_kernel<<<1, 256, 0, stream>>>(U4, bias_h1, g_h1, bln_h1, h1_f);
  ln_relu_kernel<<<1, 256, 0, stream>>>(U5, bias_h2, g_h2, bln_h2, h2_f);

  // Output GEMVs + tanh(LN(.))
  gemv_wmma_kernel<<<ggrid, gblk, 0, stream>>>(W_H1_O, h1_f, U6);
  gemv_wmma_kernel<<<ggrid, gblk, 0, stream>>>(W_H2_O, h2_f, U7);
  final_kernel<<<1, 256, 0, stream>>>(U6, U7, bias_o, g_o, bln_o, (float*)d_out);
}